// RoiPoolingConv_62577673503241
// MI455X (gfx1250) — compile-verified
//
#include <hip/hip_runtime.h>
#include <cmath>

typedef float v4f __attribute__((ext_vector_type(4)));
typedef int   v4i __attribute__((ext_vector_type(4)));

// ---- gfx1250 async global->LDS path (ASYNCcnt), guarded so we always compile.
#if defined(__HIP_DEVICE_COMPILE__) && __has_builtin(__builtin_amdgcn_global_load_async_to_lds_b128)
#define ROI_ASYNC 1
#else
#define ROI_ASYNC 0
#endif

// Builtin signature (from round-1 diagnostic): arg0 is 'v4i __device__ *'
// i.e. addrspace(1) pointer to <4 x i32>; arg1 is the LDS-side equivalent.
typedef __attribute__((address_space(1))) v4i* as1v4ip;
typedef __attribute__((address_space(3))) v4i* as3v4ip;

__device__ __forceinline__ void roi_async_b128(const float* g, float* l) {
#if ROI_ASYNC
  // Generic->AS casts must go through an integer; LDS flat addr low 32 bits
  // equal the LDS offset (aperture mapping), global flat == AS1 value.
  as1v4ip gp = (as1v4ip)(unsigned long long)g;
  as3v4ip lp = (as3v4ip)(unsigned int)(unsigned long long)l;
  __builtin_amdgcn_global_load_async_to_lds_b128(gp, lp, 0, 0);
#else
  (void)g; (void)l;
#endif
}

__device__ __forceinline__ void roi_wait_async_le4() {
#if ROI_ASYNC
#if __has_builtin(__builtin_amdgcn_s_wait_asynccnt)
  __builtin_amdgcn_s_wait_asynccnt(4);
#else
  asm volatile("s_wait_asynccnt 0x4" ::: "memory");
#endif
  asm volatile("" ::: "memory");  // keep LDS reads below the wait
#endif
}

__device__ __forceinline__ void roi_wait_async_le0() {
#if ROI_ASYNC
#if __has_builtin(__builtin_amdgcn_s_wait_asynccnt)
  __builtin_amdgcn_s_wait_asynccnt(0);
#else
  asm volatile("s_wait_asynccnt 0x0" ::: "memory");
#endif
  asm volatile("" ::: "memory");
#endif
}

__device__ __forceinline__ void blend_store(float* dst, v4f v00, v4f v01,
                                            v4f v10, v4f v11, float fx, float fy) {
  const float gx = 1.0f - fx;
  const float gy = 1.0f - fy;
  v4f top = v00 * gx + v01 * fx;   // matches reference association
  v4f bot = v10 * gx + v11 * fx;
  v4f res = top * gy + bot * fy;
  // Output is write-once streaming (103MB): NT store keeps the 32MB image
  // resident in the 192MB L2 instead of being flushed by the write stream.
  __builtin_nontemporal_store(res, (v4f*)dst);
}

// One block per (roi, oy) output row; blockDim.x = C/4 lanes (float4 per lane).
__global__ __launch_bounds__(128) void roi_bilinear_kernel(
    const float* __restrict__ image,  // [H, W, C]
    const int*   __restrict__ rois,   // [R, 4] = (x, y, w, h)
    float*       __restrict__ out,    // [R, pool, pool, C]
    int W, int C, int pool) {
  extern __shared__ float lds[];  // [2][4][C] double-buffered corner staging

  const int r   = blockIdx.x / pool;
  const int oy  = blockIdx.x - r * pool;
  const int tid = threadIdx.x;
  const int c4  = tid << 2;

  const int rx = rois[4 * r + 0];
  const int ry = rois[4 * r + 1];
  const int rw = rois[4 * r + 2];
  const int rh = rois[4 * r + 3];

  const float pf = (float)pool;

  // y-axis coords (block-uniform)
  const float hf = (float)rh;
  float sy = (oy + 0.5f) * (hf / pf) - 0.5f;
  sy = fminf(fmaxf(sy, 0.0f), hf - 1.0f);
  const int   iy0 = (int)floorf(sy);
  const int   iy1 = (iy0 + 1 < rh) ? iy0 + 1 : rh - 1;
  const float fy  = sy - (float)iy0;

  const float wf     = (float)rw;
  const float xscale = wf / pf;

  const float* row0 = image + (size_t)(ry + iy0) * W * C;
  const float* row1 = image + (size_t)(ry + iy1) * W * C;
  float*       outp = out + (size_t)blockIdx.x * pool * C + c4;  // (r*pool+oy)*pool*C

  auto xcoords = [&](int ox, int& x0, int& x1, float& fx) {
    float sx = (ox + 0.5f) * xscale - 0.5f;
    sx = fminf(fmaxf(sx, 0.0f), wf - 1.0f);
    int i0 = (int)floorf(sx);
    x1 = rx + ((i0 + 1 < rw) ? i0 + 1 : rw - 1);
    x0 = rx + i0;
    fx = sx - (float)i0;
  };

#if ROI_ASYNC
  auto issue = [&](int buf, int x0, int x1) {
    float* lb = lds + (size_t)(buf * 4) * C + c4;
    roi_async_b128(row0 + (size_t)x0 * C + c4, lb + 0 * C);
    roi_async_b128(row0 + (size_t)x1 * C + c4, lb + 1 * C);
    roi_async_b128(row1 + (size_t)x0 * C + c4, lb + 2 * C);
    roi_async_b128(row1 + (size_t)x1 * C + c4, lb + 3 * C);
  };

  int x0c, x1c; float fxc;
  xcoords(0, x0c, x1c, fxc);
  issue(0, x0c, x1c);
  for (int ox = 0; ox < pool; ++ox) {
    const int buf = ox & 1;
    float fxn = 0.0f;
    if (ox + 1 < pool) {           // prefetch next pixel's corners, then retire
      int x0n, x1n;                // the current 4 (in-order completion).
      xcoords(ox + 1, x0n, x1n, fxn);
      issue(buf ^ 1, x0n, x1n);
      roi_wait_async_le4();
    } else {
      roi_wait_async_le0();
    }
    const float* lb = lds + (size_t)(buf * 4) * C + c4;
    v4f v00 = *(const v4f*)(lb + 0 * C);
    v4f v01 = *(const v4f*)(lb + 1 * C);
    v4f v10 = *(const v4f*)(lb + 2 * C);
    v4f v11 = *(const v4f*)(lb + 3 * C);
    blend_store(outp + (size_t)ox * C, v00, v01, v10, v11, fxc, fy);
    fxc = fxn;
  }
#else
  for (int ox = 0; ox < pool; ++ox) {
    int x0, x1; float fx;
    xcoords(ox, x0, x1, fx);
    v4f v00 = *(const v4f*)(row0 + (size_t)x0 * C + c4);
    v4f v01 = *(const v4f*)(row0 + (size_t)x1 * C + c4);
    v4f v10 = *(const v4f*)(row1 + (size_t)x0 * C + c4);
    v4f v11 = *(const v4f*)(row1 + (size_t)x1 * C + c4);
    blend_store(outp + (size_t)ox * C, v00, v01, v10, v11, fx, fy);
  }
#endif
}

extern "C" void kernel_launch(void* const* d_in, const int* in_sizes, int n_in,
                              void* d_out, int out_size, void* d_ws, size_t ws_size,
                              hipStream_t stream) {
  (void)n_in; (void)d_ws; (void)ws_size;
  const float* image = (const float*)d_in[0];
  const int*   rois  = (const int*)d_in[1];
  float*       out   = (float*)d_out;

  const int H = 128, W = 128;               // per reference setup
  const int R = in_sizes[1] / 4;            // rois: [1, R, 4]
  const int C = in_sizes[0] / (H * W);      // image: [1, H, W, C]
  // pool_size lives on device (scalar input); recover it from out_size:
  // out_size = R * pool^2 * C
  const int pp   = out_size / (R * C);
  const int pool = (int)(sqrt((double)pp) + 0.5);

  dim3 block(C / 4);                        // 128 threads = 4 wave32 waves
  dim3 grid(R * pool);                      // one block per output row
  size_t shmem = (size_t)2 * 4 * C * sizeof(float);  // 16KB double buffer
  roi_bilinear_kernel<<<grid, block, shmem, stream>>>(image, rois, out, W, C, pool);
}